// Mutual_Attention_20452634263767
// MI455X (gfx1250) — compile-verified
//
#include <hip/hip_runtime.h>
#include <hip/hip_bf16.h>

typedef __attribute__((ext_vector_type(2))) float v2f;
typedef __attribute__((ext_vector_type(4))) float v4f;
typedef __attribute__((ext_vector_type(8))) float v8f;

#define BS  4
#define CH  32
#define NF  128
#define L   4096      // hei*wei
#define ROWS 4096     // BS*CH*HEI

__device__ __forceinline__ v8f wmma_f32(v2f a, v2f b, v8f c) {
  // V_WMMA_F32_16X16X4_F32 : D = A(16x4) * B(4x16) + C(16x16)
  return __builtin_amdgcn_wmma_f32_16x16x4_f32(false, a, false, b, (short)0, c,
                                               false, false);
}

// ---------------------------------------------------------------------------
// Kernel A: affinity/alpha  (1x1 conv -> BN(eval) -> ReLU -> 2-way softmax)
// ---------------------------------------------------------------------------
__global__ void alpha_kernel(const float* __restrict__ head,
                             const float* __restrict__ tail,
                             const float* __restrict__ conv_w,
                             const float* __restrict__ conv_b,
                             const float* __restrict__ bn_w,
                             const float* __restrict__ bn_b,
                             const float* __restrict__ bn_mean,
                             const float* __restrict__ bn_var,
                             const int* __restrict__ flagp,
                             float* __restrict__ alpha) {
  int t = blockIdx.x * blockDim.x + threadIdx.x;
  if (t >= BS * L) return;
  int b = t >> 12, l = t & (L - 1);
  const float* hb = head + ((b * CH) << 12) + l;
  const float* tb = tail + ((b * CH) << 12) + l;
  float s0 = conv_b[0], s1 = conv_b[1];
#pragma unroll 4
  for (int c = 0; c < CH; ++c) {
    float hv = hb[c << 12];
    s0 = fmaf(conv_w[c], hv, s0);
    s1 = fmaf(conv_w[64 + c], hv, s1);
  }
#pragma unroll 4
  for (int c = 0; c < CH; ++c) {
    float tv = tb[c << 12];
    s0 = fmaf(conv_w[CH + c], tv, s0);
    s1 = fmaf(conv_w[64 + CH + c], tv, s1);
  }
  float sc0 = bn_w[0] * rsqrtf(bn_var[0] + 1e-5f);
  float sc1 = bn_w[1] * rsqrtf(bn_var[1] + 1e-5f);
  float r0 = fmaxf((s0 - bn_mean[0]) * sc0 + bn_b[0], 0.f);
  float r1 = fmaxf((s1 - bn_mean[1]) * sc1 + bn_b[1], 0.f);
  float m = fmaxf(r0, r1);
  float e0 = __expf(r0 - m), e1 = __expf(r1 - m);
  alpha[t] = (*flagp) ? (e0 / (e0 + e1)) : (e1 / (e0 + e1));
}

// ---------------------------------------------------------------------------
// Kernel B: five projections (4096x128 @ 128x128) via WMMA f32 16x16x4,
// scattering into attention-friendly layouts:
//   q  [b][l][32]          K' [b][l][64] (cols 0..31 pre-scaled by alpha)
//   V' [b][64][l]
// One wave computes one 16x16 output tile.
// ---------------------------------------------------------------------------
__global__ __launch_bounds__(128) void proj_kernel(
    const float* __restrict__ head, const float* __restrict__ tail,
    const float* __restrict__ R_q, const float* __restrict__ R_k,
    const float* __restrict__ R_v, const float* __restrict__ F_q,
    const float* __restrict__ F_k, const float* __restrict__ F_v,
    const int* __restrict__ flagp, const float* __restrict__ alpha,
    float* __restrict__ qws, float* __restrict__ kc, float* __restrict__ vc) {
  int wid = (blockIdx.x * blockDim.x + threadIdx.x) >> 5;
  int lane = threadIdx.x & 31;
  int half = lane >> 4, lm = lane & 15;
  int proj = wid >> 11;            // 0..4 (2048 tiles each)
  int t = wid & 2047;
  int r0 = (t >> 3) << 4;          // 256 row tiles
  int n0 = (t & 7) << 4;           // 8 col tiles
  int flag = *flagp;

  const float* src;
  const float* mat;
  switch (proj) {
    case 0:  src = flag ? head : tail; mat = flag ? R_q : F_q; break; // q
    case 1:  src = flag ? tail : head; mat = flag ? F_k : R_k; break; // K' lo (scaled)
    case 2:  src = flag ? head : tail; mat = flag ? R_k : F_k; break; // K' hi
    case 3:  src = flag ? tail : head; mat = flag ? F_v : R_v; break; // V' rows 0..31
    default: src = flag ? head : tail; mat = flag ? R_v : F_v; break; // V' rows 32..63
  }

  const float* arow = src + (r0 + lm) * NF;
  v8f acc = {};
#pragma unroll
  for (int k0 = 0; k0 < NF; k0 += 4) {
    int k = k0 + 2 * half;
    v2f a;  a.x = arow[k];                a.y = arow[k + 1];
    v2f bv; bv.x = mat[k * NF + n0 + lm]; bv.y = mat[(k + 1) * NF + n0 + lm];
    acc = wmma_f32(a, bv, acc);
  }

  int w = n0 + lm;
#pragma unroll
  for (int r = 0; r < 8; ++r) {
    int row = r0 + half * 8 + r;             // (b, c, h) flattened
    int b = row >> 10, c = (row >> 5) & 31, h = row & 31;
    int l = (h << 7) + w;
    int bl = (b << 12) + l;
    float v = acc[r];
    if (proj == 0)      qws[bl * 32 + c] = v;
    else if (proj == 1) kc[bl * 64 + c] = v * alpha[bl];
    else if (proj == 2) kc[bl * 64 + 32 + c] = v;
    else if (proj == 3) vc[((b * 64 + c) << 12) + l] = v;
    else                vc[((b * 64 + 32 + c) << 12) + l] = v;
  }
}

// ---------------------------------------------------------------------------
// Kernel C: fused flash attention, 64-column tiles.
//   S = K'(L x 64) @ V'(64 x L), online softmax, O += P @ q(L x 32)
// 8 waves/block, 16 rows/wave; V'/q tiles staged in LDS with float4 loads;
// one softmax pass (one butterfly + one correction) per 64 columns.
// ---------------------------------------------------------------------------
__global__ __launch_bounds__(256, 1) void attn_kernel(
    const float* __restrict__ qws, const float* __restrict__ kc,
    const float* __restrict__ vc, float* __restrict__ y) {
  __shared__ float lds_v[64][64];      // V' column tile   (16 KB)
  __shared__ float lds_q[64][32];      // q row tile       (8 KB)
  __shared__ float lds_p[8][16][66];   // per-wave P tiles (33 KB, padded)

  int tid = threadIdx.x;
  int w = tid >> 5;
  int lane = tid & 31;
  int half = lane >> 4, lm = lane & 15;
  int b = blockIdx.x >> 5;                         // 32 blocks per batch
  int i0 = (((blockIdx.x & 31) << 3) + w) << 4;    // wave's first row
  int bl = b << 12;

  const float* vcB = vc + ((b * 64) << 12);
  const float* qB = qws + bl * 32;
  const float* kcRow = kc + (bl + i0 + lm) * 64;

  // Preload this wave's 16x64 K' strip in WMMA A-layout (16 k-steps).
  v2f ka[16];
#pragma unroll
  for (int s = 0; s < 16; ++s) {
    int k0 = 4 * s + 2 * half;
    ka[s].x = kcRow[k0];
    ka[s].y = kcRow[k0 + 1];
  }

  v8f o0 = {}, o1 = {};
  float mrow[8], lrow[8];
#pragma unroll
  for (int r = 0; r < 8; ++r) { mrow[r] = -3.0e38f; lrow[r] = 0.f; }

  for (int j0 = 0; j0 < L; j0 += 64) {
    __syncthreads();   // previous iteration's LDS reads complete
    // Cooperative float4 stage: V' tile (64x64) and q tile (64x32)
#pragma unroll
    for (int i = 0; i < 4; ++i) {
      int idx = (tid + (i << 8)) << 2;       // 0..4092
      int kr = idx >> 6, cn = idx & 63;
      *(v4f*)&lds_v[kr][cn] = *(const v4f*)&vcB[(kr << 12) + j0 + cn];
    }
#pragma unroll
    for (int i = 0; i < 2; ++i) {
      int idx = (tid + (i << 8)) << 2;       // 0..2044
      int kr = idx >> 5, cc = idx & 31;
      *(v4f*)&lds_q[kr][cc] = *(const v4f*)&qB[(j0 + kr) * 32 + cc];
    }
    __syncthreads();

    // Score tiles: 4 x (16x16), K = 64
    v8f s0 = {}, s1 = {}, s2 = {}, s3 = {};
#pragma unroll
    for (int st = 0; st < 16; ++st) {
      int k0 = 4 * st + 2 * half;
      v2f b0; b0.x = lds_v[k0][lm];      b0.y = lds_v[k0 + 1][lm];
      v2f b1; b1.x = lds_v[k0][16 + lm]; b1.y = lds_v[k0 + 1][16 + lm];
      v2f b2; b2.x = lds_v[k0][32 + lm]; b2.y = lds_v[k0 + 1][32 + lm];
      v2f b3; b3.x = lds_v[k0][48 + lm]; b3.y = lds_v[k0 + 1][48 + lm];
      s0 = wmma_f32(ka[st], b0, s0);
      s1 = wmma_f32(ka[st], b1, s1);
      s2 = wmma_f32(ka[st], b2, s2);
      s3 = wmma_f32(ka[st], b3, s3);
    }

    // Online softmax over the 64-column strip; row stats in-lane per half-wave.
#pragma unroll
    for (int r = 0; r < 8; ++r) {
      float v = fmaxf(fmaxf(s0[r], s1[r]), fmaxf(s2[r], s3[r]));
      v = fmaxf(v, __shfl_xor(v, 1, 16));
      v = fmaxf(v, __shfl_xor(v, 2, 16));
      v = fmaxf(v, __shfl_xor(v, 4, 16));
      v = fmaxf(v, __shfl_xor(v, 8, 16));
      float mnew = fmaxf(mrow[r], v);
      float corr = __expf(mrow[r] - mnew);
      mrow[r] = mnew;
      int m = half * 8 + r;
      float p0 = __expf(s0[r] - mnew);
      float p1 = __expf(s1[r] - mnew);
      float p2 = __expf(s2[r] - mnew);
      float p3 = __expf(s3[r] - mnew);
      lds_p[w][m][lm] = p0;
      lds_p[w][m][16 + lm] = p1;
      lds_p[w][m][32 + lm] = p2;
      lds_p[w][m][48 + lm] = p3;
      float ps = (p0 + p1) + (p2 + p3);
      ps += __shfl_xor(ps, 1, 16);
      ps += __shfl_xor(ps, 2, 16);
      ps += __shfl_xor(ps, 4, 16);
      ps += __shfl_xor(ps, 8, 16);
      lrow[r] = lrow[r] * corr + ps;
      o0[r] *= corr;
      o1[r] *= corr;
    }
    __syncthreads();

    // O += P(16x64) @ q(64x32), K = 64
#pragma unroll
    for (int st = 0; st < 16; ++st) {
      int k0 = 4 * st + 2 * half;
      v2f pa;  pa.x = lds_p[w][lm][k0];    pa.y = lds_p[w][lm][k0 + 1];
      v2f qb0; qb0.x = lds_q[k0][lm];      qb0.y = lds_q[k0 + 1][lm];
      v2f qb1; qb1.x = lds_q[k0][16 + lm]; qb1.y = lds_q[k0 + 1][16 + lm];
      o0 = wmma_f32(pa, qb0, o0);
      o1 = wmma_f32(pa, qb1, o1);
    }
  }

  // Normalize and scatter to y[b][c][l]
#pragma unroll
  for (int r = 0; r < 8; ++r) {
    int m = half * 8 + r;
    float inv = 1.f / lrow[r];
    y[((b * 32 + lm) << 12) + i0 + m] = o0[r] * inv;
    y[((b * 32 + 16 + lm) << 12) + i0 + m] = o1[r] * inv;
  }
}

// ---------------------------------------------------------------------------
// Kernel D: out = y @ W + residual   (4096x128 @ 128x128, WMMA tiles)
// ---------------------------------------------------------------------------
__global__ __launch_bounds__(128) void final_kernel(
    const float* __restrict__ y, const float* __restrict__ head,
    const float* __restrict__ tail, const float* __restrict__ R_W,
    const float* __restrict__ F_W, const int* __restrict__ flagp,
    float* __restrict__ out) {
  int wid = (blockIdx.x * blockDim.x + threadIdx.x) >> 5;
  int lane = threadIdx.x & 31;
  int half = lane >> 4, lm = lane & 15;
  int r0 = (wid >> 3) << 4;
  int n0 = (wid & 7) << 4;
  int flag = *flagp;
  const float* mat = flag ? R_W : F_W;
  const float* res = flag ? head : tail;

  const float* arow = y + (r0 + lm) * NF;   // y is row-major 4096x128
  v8f acc = {};
#pragma unroll
  for (int k0 = 0; k0 < NF; k0 += 4) {
    int k = k0 + 2 * half;
    v2f a;  a.x = arow[k];                a.y = arow[k + 1];
    v2f bv; bv.x = mat[k * NF + n0 + lm]; bv.y = mat[(k + 1) * NF + n0 + lm];
    acc = wmma_f32(a, bv, acc);
  }
#pragma unroll
  for (int r = 0; r < 8; ++r) {
    int row = r0 + half * 8 + r;
    int idx = row * NF + n0 + lm;
    out[idx] = acc[r] + res[idx];
  }
}

// ---------------------------------------------------------------------------
extern "C" void kernel_launch(void* const* d_in, const int* in_sizes, int n_in,
                              void* d_out, int out_size, void* d_ws,
                              size_t ws_size, hipStream_t stream) {
  const float* head    = (const float*)d_in[0];
  const float* tail    = (const float*)d_in[1];
  const float* R_q     = (const float*)d_in[2];
  const float* R_k     = (const float*)d_in[3];
  const float* R_v     = (const float*)d_in[4];
  const float* R_W     = (const float*)d_in[5];
  const float* F_q     = (const float*)d_in[6];
  const float* F_k     = (const float*)d_in[7];
  const float* F_v     = (const float*)d_in[8];
  const float* F_W     = (const float*)d_in[9];
  const float* conv_w  = (const float*)d_in[10];
  const float* conv_b  = (const float*)d_in[11];
  const float* bn_w    = (const float*)d_in[12];
  const float* bn_b    = (const float*)d_in[13];
  const float* bn_mean = (const float*)d_in[14];
  const float* bn_var  = (const float*)d_in[15];
  const int*   flag    = (const int*)d_in[16];
  float* out = (float*)d_out;

  float* ws    = (float*)d_ws;
  float* alpha = ws;                 // 16384
  float* qws   = alpha + 16384;      // 524288   q  [b][l][32]
  float* kc    = qws + 524288;       // 1048576  K' [b][l][64]
  float* vc    = kc + 1048576;       // 1048576  V' [b][64][l]
  float* y     = vc + 1048576;       // 524288   y  [b][32][l]

  alpha_kernel<<<64, 256, 0, stream>>>(head, tail, conv_w, conv_b, bn_w, bn_b,
                                       bn_mean, bn_var, flag, alpha);
  proj_kernel<<<2560, 128, 0, stream>>>(head, tail, R_q, R_k, R_v, F_q, F_k,
                                        F_v, flag, alpha, qws, kc, vc);
  attn_kernel<<<128, 256, 0, stream>>>(qws, kc, vc, y);
  final_kernel<<<512, 128, 0, stream>>>(y, head, tail, R_W, F_W, flag, out);
}